// CTCLoss_30047591203274
// MI455X (gfx1250) — compile-verified
//
#include <hip/hip_runtime.h>
#include <math.h>

// ---------------------------------------------------------------------------
// CTC loss for B=128, T=128, C=6625, L=25 (S = 2L+1 = 51), float32.
// Bandwidth-bound: 434 MB input read exactly once (~19us floor @ 23.3 TB/s).
// CDNA5 path: async global->LDS staging (b128 bulk, ASYNCcnt) with the 24-bit
// instruction offset applied to both LDS and global sides => zero per-issue
// address math. Compute (max / sum-exp / gather) runs out of LDS.
// ---------------------------------------------------------------------------

#define NEGV (-1e30f)

constexpr int Bc = 128;
constexpr int Tc = 128;
constexpr int Cc = 6625;
constexpr int Lc = 25;
constexpr int Sc = 2 * Lc + 1; // 51

// ---------------------------------------------------------------------------
// Phase 1: per-(b,t) row logsumexp + gather of extended-label emissions.
// One 256-thread block (8 wave32) per row.
// ---------------------------------------------------------------------------
__global__ __launch_bounds__(256) void lse_gather_kernel(
    const float* __restrict__ predicts, // [B,T,C]
    const int*   __restrict__ labels,   // [B,L]
    float*       __restrict__ emit)     // [B,T,S]
{
    // 26,500 B row + 16 B slack for the runtime alignment pad.
    __shared__ __align__(16) float rowbuf[Cc + 4];
    __shared__ float red[8];
    __shared__ float bcast0;
    __shared__ float bcast1;

    const int bt  = blockIdx.x;          // bt = b*T + t
    const int b   = bt / Tc;
    const int tid = threadIdx.x;

    const char* srcb = (const char*)(predicts + (size_t)bt * Cc);

    // Row base is only 4-B aligned (C*4 % 16 != 0). Peel 'lead' elements so
    // the bulk is 16-B aligned on BOTH the global and LDS side.
    const unsigned mis  = (unsigned)((size_t)srcb & 15); // 0,4,8,12
    const int lead  = (int)(((16u - mis) & 15u) >> 2);   // 0..3 elements
    const int pad   = (16 - 4 * lead) & 15;              // LDS byte pad (0..12)
    float* row = (float*)((char*)rowbuf + pad);          // row[i] == element i

    const int nvec  = (Cc - lead) >> 2;                  // full float4s (>=1655)
    const int tailn = (Cc - lead) & 3;                   // 0..3 trailing floats
    const int tail0 = lead + 4 * nvec;

    // Per-lane base addresses for the aligned bulk (computed once).
    const unsigned lds_base =
        (unsigned)(unsigned long long)(const void*)row;  // low 32 bits = LDS offset
    const unsigned lds_v = lds_base + (unsigned)(4 * lead + 16 * tid);
    const unsigned long long g_v =
        (unsigned long long)(const void*)(srcb + 4 * lead + 16 * tid);

    // 256 lanes x 16 B = 4096 B per issue; offset applies to LDS and global.
#define ISSUE_B128(OFF)                                                       \
    asm volatile("global_load_async_to_lds_b128 %0, %1, off offset:" #OFF     \
                 :: "v"(lds_v), "v"(g_v) : "memory")
    ISSUE_B128(0);
    ISSUE_B128(4096);
    ISSUE_B128(8192);
    ISSUE_B128(12288);
    ISSUE_B128(16384);
    ISSUE_B128(20480);                 // covers vec idx 1280+tid <= 1535
    if (tid < nvec - 1536) {           // guarded 7th issue (119 or 120 lanes)
        ISSUE_B128(24576);
    }
#undef ISSUE_B128

    // Unaligned lead / tail elements via b32 (4-B aligned always legal).
    if (tid < lead) {
        const unsigned la = lds_base + (unsigned)(4 * tid);
        const unsigned long long ga = (unsigned long long)(const void*)(srcb + 4 * tid);
        asm volatile("global_load_async_to_lds_b32 %0, %1, off"
                     :: "v"(la), "v"(ga) : "memory");
    }
    if (tid < tailn) {
        const int i = tail0 + tid;
        const unsigned la = lds_base + (unsigned)(4 * i);
        const unsigned long long ga = (unsigned long long)(const void*)(srcb + 4 * i);
        asm volatile("global_load_async_to_lds_b32 %0, %1, off"
                     :: "v"(la), "v"(ga) : "memory");
    }

#if __has_builtin(__builtin_amdgcn_s_wait_asynccnt)
    __builtin_amdgcn_s_wait_asynccnt(0);
#else
    asm volatile("s_wait_asynccnt 0" ::: "memory");
#endif
    __syncthreads();

    // ---- pass 1 (LDS): row max. Constant-trip, fully unrolled (26 iters).
    float lmax = NEGV;
#pragma unroll
    for (int j = 0; j < 26; ++j) {
        const int i = tid + j * 256;
        if (j < 25 || i < Cc) lmax = fmaxf(lmax, row[i]);
    }
    for (int m = 16; m > 0; m >>= 1)
        lmax = fmaxf(lmax, __shfl_xor(lmax, m, 32));      // wave32
    const int wid = tid >> 5;
    if ((tid & 31) == 0) red[wid] = lmax;
    __syncthreads();
    if (tid == 0) {
        float mx = red[0];
        for (int w = 1; w < 8; ++w) mx = fmaxf(mx, red[w]);
        bcast0 = mx;
    }
    __syncthreads();
    const float rowmax = bcast0;

    // ---- pass 2 (LDS): sum of exp(x - max)
    float lsum = 0.0f;
#pragma unroll
    for (int j = 0; j < 26; ++j) {
        const int i = tid + j * 256;
        if (j < 25 || i < Cc) lsum += __expf(row[i] - rowmax);
    }
    for (int m = 16; m > 0; m >>= 1)
        lsum += __shfl_xor(lsum, m, 32);
    if ((tid & 31) == 0) red[wid] = lsum;
    __syncthreads();
    if (tid == 0) {
        float s = 0.0f;
        for (int w = 0; w < 8; ++w) s += red[w];
        bcast1 = rowmax + __logf(s);
    }
    __syncthreads();
    const float lse = bcast1;

    // ---- gather emissions at extended labels (blank interleaved, blank=0)
    if (tid < Sc) {
        const int extc = (tid & 1) ? labels[b * Lc + (tid >> 1)] : 0;
        emit[(size_t)bt * Sc + tid] = row[extc] - lse;
    }
}

// ---------------------------------------------------------------------------
// Phase 2: alpha recursion. One 64-thread block per batch element; state in
// LDS, T=128 sequential steps. Deterministic (fixed order, no atomics).
// ---------------------------------------------------------------------------
__global__ __launch_bounds__(64) void ctc_alpha_kernel(
    const float* __restrict__ emit,      // [B,T,S]
    const int*   __restrict__ labels,    // [B,L]
    const int*   __restrict__ lens,      // [B]
    float*       __restrict__ loss_out)  // [B]
{
    const int b = blockIdx.x;
    const int s = threadIdx.x;
    __shared__ float alpha[64];
    __shared__ int   ext[64];

    const bool active = (s < Sc);
    if (active) ext[s] = (s & 1) ? labels[b * Lc + (s >> 1)] : 0;
    __syncthreads();
    const bool skip = active && (s >= 2) && (ext[s] != ext[s - 2]);

    const float* em = emit + (size_t)b * Tc * Sc;

    float a = NEGV;
    if (s == 0) a = em[0];
    else if (s == 1) a = em[1];
    alpha[s] = a;
    __syncthreads();

    for (int t = 1; t < Tc; ++t) {
        const float a0 = alpha[s];
        const float a1 = (s >= 1) ? alpha[s - 1] : NEGV;
        const float a2 = skip ? alpha[s - 2] : NEGV;
        float nv = NEGV;
        if (active) {
            const float m = fmaxf(a0, fmaxf(a1, a2));
            nv = m + logf(expf(a0 - m) + expf(a1 - m) + expf(a2 - m))
                   + em[(size_t)t * Sc + s];
        }
        __syncthreads();
        alpha[s] = nv;
        __syncthreads();
    }

    if (s == 0) {
        const int   len = lens[b];
        const int   idx = 2 * len;               // final blank position
        const float x = alpha[idx], y = alpha[idx - 1];
        const float m = fmaxf(x, y);
        const float ll = m + logf(expf(x - m) + expf(y - m));
        float l = -ll;
        if (l > 1e29f) l = 0.0f;                 // zero_infinity
        loss_out[b] = l / (float)len;            // per-sample /target_length
    }
}

// ---------------------------------------------------------------------------
// Phase 3: mean over batch, then the module's extra /B (=> sum / (B*B)).
// ---------------------------------------------------------------------------
__global__ __launch_bounds__(128) void finalize_kernel(
    const float* __restrict__ loss_b, float* __restrict__ out)
{
    __shared__ float part[4];
    const int tid = threadIdx.x;
    float v = loss_b[tid];
    for (int m = 16; m > 0; m >>= 1) v += __shfl_xor(v, m, 32);
    if ((tid & 31) == 0) part[tid >> 5] = v;
    __syncthreads();
    if (tid == 0)
        out[0] = (part[0] + part[1] + part[2] + part[3]) *
                 (1.0f / ((float)Bc * (float)Bc));
}

// ---------------------------------------------------------------------------
extern "C" void kernel_launch(void* const* d_in, const int* in_sizes, int n_in,
                              void* d_out, int out_size, void* d_ws, size_t ws_size,
                              hipStream_t stream)
{
    (void)in_sizes; (void)n_in; (void)out_size; (void)ws_size;
    const float* predicts = (const float*)d_in[0]; // [B,T,C] f32
    const int*   labels   = (const int*)d_in[1];   // [B,L]   i32
    const int*   lens     = (const int*)d_in[2];   // [B]     i32
    float*       out      = (float*)d_out;         // scalar f32

    float* emit   = (float*)d_ws;                            // B*T*S floats (~3.34 MB)
    float* loss_b = emit + (size_t)Bc * Tc * Sc;             // B floats

    lse_gather_kernel<<<Bc * Tc, 256, 0, stream>>>(predicts, labels, emit);
    ctc_alpha_kernel <<<Bc,      64,  0, stream>>>(emit, labels, lens, loss_b);
    finalize_kernel  <<<1,       128, 0, stream>>>(loss_b, out);
}